// YOLOLoss_60894046323287
// MI455X (gfx1250) — compile-verified
//
#include <hip/hip_runtime.h>
#include <hip/hip_bf16.h>
#include <math.h>

// ---------------------------------------------------------------------------
// YOLO loss for MI455X (gfx1250, wave32).
// Two launches: per-batch partial sums (gather + transcendental work), then a
// single-wave finalizer that reduces the 64x5 partial matrix with
// V_WMMA_F32_16X16X4_F32 and applies the normalization.
// All reductions are fixed-order (shuffle/LDS trees, no float atomics) so the
// result is bit-identical across graph replays.
// ---------------------------------------------------------------------------

#define NUM_CLASSES 4
#define IMG_SIZE    640.0f
#define T_PER_B     50
#define NEG_PER_B   500
#define BATCH       64
#define EPSF        1e-7f

typedef __attribute__((ext_vector_type(2))) float v2f;
typedef __attribute__((ext_vector_type(8))) float v8f;

__device__ __forceinline__ float sigmoidf_(float x) {
    return 1.0f / (1.0f + __expf(-x) * 0.0f + expf(-x)); // keep precise expf
}
__device__ __forceinline__ float sigmoid_precise(float x) {
    return 1.0f / (1.0f + expf(-x));
}
// jax.nn.softplus(x) == max(x,0) + log1p(exp(-|x|))  (numerically stable)
__device__ __forceinline__ float softplusf_(float x) {
    return fmaxf(x, 0.0f) + log1pf(expf(-fabsf(x)));
}

__global__ __launch_bounds__(256)
void yolo_partials(const float* __restrict__ p3,
                   const float* __restrict__ p4,
                   const float* __restrict__ p5,
                   const float* __restrict__ targets,
                   const float* __restrict__ anchors,
                   const int*   __restrict__ neg3,
                   const int*   __restrict__ neg4,
                   const int*   __restrict__ neg5,
                   float*       __restrict__ ws)
{
    const int b   = blockIdx.x;
    const int tid = threadIdx.x;

    const float* preds[3] = { p3, p4, p5 };
    const int*   negs[3]  = { neg3, neg4, neg5 };
    const int    dims[3]  = { 80, 40, 20 };

    float box = 0.0f, cls = 0.0f, conf = 0.0f, nvalid = 0.0f, neg = 0.0f;

    // ---------------- positive targets: 3 scales x 50 targets --------------
    if (tid < 3 * T_PER_B) {
        const int s   = tid / T_PER_B;
        const int t   = tid - s * T_PER_B;
        const int dim = dims[s];
        const int hw  = dim * dim;
        const float fdim = (float)dim;

        const float* tg = targets + ((size_t)b * T_PER_B + t) * 5;
        const float clsf = tg[0];
        const float tx = tg[1], ty = tg[2], tw = tg[3], th = tg[4];
        const int cls_id = (int)clsf;

        const bool valid = (tx >= 0.0f) && (tx <= 1.0f) &&
                           (ty >= 0.0f) && (ty <= 1.0f) &&
                           (tw >  0.0f) && (tw <= 1.0f) &&
                           (th >  0.0f) && (th <= 1.0f);

        // anchor sizes (normalized), best-anchor argmax (first max wins)
        float aw[3], ah[3];
        #pragma unroll
        for (int a = 0; a < 3; ++a) {
            aw[a] = anchors[(3 * s + a) * 2 + 0] / IMG_SIZE;
            ah[a] = anchors[(3 * s + a) * 2 + 1] / IMG_SIZE;
        }
        int best = 0; float best_r = -3.0e38f;
        #pragma unroll
        for (int a = 0; a < 3; ++a) {
            const float inter = fminf(tw, aw[a]) * fminf(th, ah[a]);
            const float uni   = tw * th + aw[a] * ah[a] - inter;
            const float r     = inter / (uni + EPSF);
            if (r > best_r) { best_r = r; best = a; }
        }

        int gxi = (int)floorf(tx * fdim); gxi = min(max(gxi, 0), dim - 1);
        int gyi = (int)floorf(ty * fdim); gyi = min(max(gyi, 0), dim - 1);

        const float* base = preds[s] +
            ((size_t)b * 27 + (size_t)best * 9) * (size_t)hw +
            (size_t)(gyi * dim + gxi);

        const float v0 = base[0];
        const float v1 = base[(size_t)hw];
        const float v2 = base[(size_t)2 * hw];
        const float v3 = base[(size_t)3 * hw];
        const float v4 = base[(size_t)4 * hw];
        const float v5 = base[(size_t)5 * hw];
        const float v6 = base[(size_t)6 * hw];
        const float v7 = base[(size_t)7 * hw];
        const float v8 = base[(size_t)8 * hw];

        const float px = sigmoid_precise(v0) + (float)gxi / fdim;
        const float py = sigmoid_precise(v1) + (float)gyi / fdim;
        const float pw = expf(v2) * aw[best];
        const float ph = expf(v3) * ah[best];

        // GIoU
        const float x1n = px - pw * 0.5f, y1n = py - ph * 0.5f;
        const float x1x = px + pw * 0.5f, y1x = py + ph * 0.5f;
        const float x2n = tx - tw * 0.5f, y2n = ty - th * 0.5f;
        const float x2x = tx + tw * 0.5f, y2x = ty + th * 0.5f;
        const float iw = fmaxf(fminf(x1x, x2x) - fmaxf(x1n, x2n), 0.0f);
        const float ih = fmaxf(fminf(y1x, y2x) - fmaxf(y1n, y2n), 0.0f);
        const float inter2 = iw * ih;
        const float uni2   = pw * ph + tw * th - inter2 + EPSF;
        const float iou    = inter2 / uni2;
        const float ew  = fmaxf(x1x, x2x) - fminf(x1n, x2n);
        const float eh  = fmaxf(y1x, y2x) - fminf(y1n, y2n);
        const float enc = ew * eh + EPSF;
        const float giou = iou - (enc - uni2) / enc;

        const float mask = (valid && (cls_id < NUM_CLASSES)) ? 1.0f : 0.0f;

        box = mask * (1.0f - giou);

        // class BCE: mean over 4 classes of softplus(z) - onehot*z
        const int cc = min(max(cls_id, 0), NUM_CLASSES - 1);
        const float zsel = (cc == 0) ? v5 : (cc == 1) ? v6 : (cc == 2) ? v7 : v8;
        const float in_range = (cls_id >= 0 && cls_id < NUM_CLASSES) ? 1.0f : 0.0f;
        const float cls_bce = (softplusf_(v5) + softplusf_(v6) +
                               softplusf_(v7) + softplusf_(v8) -
                               in_range * zsel) * 0.25f;
        cls = mask * cls_bce;

        const float conf_t = fminf(fmaxf(giou, 0.0f), 1.0f);
        conf = mask * (softplusf_(v4) - conf_t * v4);

        nvalid = valid ? 1.0f : 0.0f;
    }

    // ---------------- negative-confidence gathers: 3 scales x 500 ----------
    for (int i = tid; i < 3 * NEG_PER_B; i += 256) {
        const int s   = i / NEG_PER_B;
        const int j   = i - s * NEG_PER_B;
        const int dim = dims[s];
        const int hw  = dim * dim;
        const int idx = negs[s][b * NEG_PER_B + j];   // in [0, 3*hw)
        const int a   = idx / hw;
        const int r   = idx - a * hw;
        const float v = preds[s][((size_t)b * 27 + (size_t)a * 9 + 4) * (size_t)hw
                                 + (size_t)r];
        neg += softplusf_(v);
    }

    // ---------------- deterministic block reduction (fixed-order tree) -----
    __shared__ float sm[5][256];
    sm[0][tid] = box; sm[1][tid] = cls; sm[2][tid] = conf;
    sm[3][tid] = nvalid; sm[4][tid] = neg;
    __syncthreads();
    for (int st = 128; st > 0; st >>= 1) {
        if (tid < st) {
            #pragma unroll
            for (int q = 0; q < 5; ++q) sm[q][tid] += sm[q][tid + st];
        }
        __syncthreads();
    }
    if (tid < 5) ws[b * 5 + tid] = sm[tid][0];
}

// ---------------------------------------------------------------------------
// Finalizer: single wave32. For each of the 5 quantities, load the 64
// per-batch partials as a 16x4 f32 A-matrix (A layout per ISA 7.12.2:
// lanes 0-15 hold K=0,1; lanes 16-31 hold K=2,3; M = lane&15) and multiply by
// an all-ones 4x16 B with V_WMMA_F32_16X16X4_F32. Every element of D row m is
// rowsum(m), so the sum of ALL 256 D elements equals 16 * total (layout-robust
// extraction), reduced with a fixed shuffle tree.
// ---------------------------------------------------------------------------
__global__ __launch_bounds__(32)
void yolo_finalize(const float* __restrict__ ws, float* __restrict__ out)
{
    const int lane = threadIdx.x;           // 0..31, EXEC all ones for WMMA
    const int m    = lane & 15;
    const int kb   = (lane < 16) ? 0 : 2;

    v2f bones; bones[0] = 1.0f; bones[1] = 1.0f;

    float sums[5];
    #pragma unroll
    for (int q = 0; q < 5; ++q) {
        v2f a;
        a[0] = ws[(m * 4 + kb + 0) * 5 + q];
        a[1] = ws[(m * 4 + kb + 1) * 5 + q];
        v8f c = {};
        // D = A x ones + 0  ->  D[m][n] = rowsum(m)
        v8f d = __builtin_amdgcn_wmma_f32_16x16x4_f32(
            /*neg_a=*/false, a, /*neg_b=*/false, bones,
            /*c_mod=*/(short)0, c, /*reuse_a=*/false, /*reuse_b=*/false);
        float t = 0.0f;
        #pragma unroll
        for (int j = 0; j < 8; ++j) t += d[j];
        #pragma unroll
        for (int off = 16; off > 0; off >>= 1)
            t += __shfl_down(t, off, 32);
        sums[q] = __shfl(t, 0, 32) * (1.0f / 16.0f);   // 16 copies of each row
    }

    if (lane == 0) {
        float box  = sums[0];
        float cls  = sums[1];
        float conf = sums[2];
        float n    = sums[3];
        float neg  = sums[4];

        const float denom = fmaxf(n, 1.0f);
        if (n > 0.0f) { box /= denom; cls /= denom; conf /= denom; }

        // each scale adds 0.5 * sum_b(mean_j softplus) / (B*500)
        //  == sum_{b,j} softplus * 0.5 / (500 * B * 500); same constant per scale
        conf += neg * (0.5f / (500.0f * (float)(BATCH * NEG_PER_B)));

        out[0] = box + cls + conf;   // total
        out[1] = box;
        out[2] = cls;
        out[3] = conf;
    }
}

extern "C" void kernel_launch(void* const* d_in, const int* in_sizes, int n_in,
                              void* d_out, int out_size, void* d_ws, size_t ws_size,
                              hipStream_t stream) {
    const float* p3      = (const float*)d_in[0];
    const float* p4      = (const float*)d_in[1];
    const float* p5      = (const float*)d_in[2];
    const float* targets = (const float*)d_in[3];
    const float* anchors = (const float*)d_in[4];
    const int*   neg3    = (const int*)d_in[5];
    const int*   neg4    = (const int*)d_in[6];
    const int*   neg5    = (const int*)d_in[7];
    float*       out     = (float*)d_out;
    float*       ws      = (float*)d_ws;   // 64 blocks x 5 partials = 1280 B

    (void)in_sizes; (void)n_in; (void)out_size; (void)ws_size;

    yolo_partials<<<BATCH, 256, 0, stream>>>(p3, p4, p5, targets, anchors,
                                             neg3, neg4, neg5, ws);
    yolo_finalize<<<1, 32, 0, stream>>>(ws, out);
}